// SparseGraphDecoder_2594160246839
// MI455X (gfx1250) — compile-verified
//
#include <hip/hip_runtime.h>
#include <hip/hip_bf16.h>
#include <math.h>

// ---------------------------------------------------------------------------
// SparseGraphDecoder for MI455X (gfx1250, wave32).
// B=4, N=4096, D=256, H1=256, H0=512, K=8.
// Gram matrix zn@zn^T (34.4 GFLOP) via V_WMMA_F32_16X16X4_F32 (fp32 matrix
// pipe preserves reference distance ordering; problem is HBM-bound so low
// precision buys nothing). B-column panels are staged in LDS, double-buffered
// and filled by the Tensor Data Mover (tensor_load_to_lds + s_wait_tensorcnt),
// with a cooperative-copy fallback when the builtin is unavailable.
// ---------------------------------------------------------------------------

typedef __attribute__((ext_vector_type(2))) float v2f;
typedef __attribute__((ext_vector_type(8))) float v8f;
typedef __attribute__((ext_vector_type(4))) unsigned int v4u;
typedef __attribute__((ext_vector_type(4))) int v4i;
typedef __attribute__((ext_vector_type(8))) int v8i;

#define B_   4
#define N_   4096
#define D_   256
#define H1_  256
#define H0_  512
#define K_   8
#define NT_  (N_ / 16)              /* 256 column tiles */
#define BST_ 260                    /* padded LDS stride (bank-conflict free) */
#define EPB_ (N_ * K_ + N_)         /* edges per batch = 36864 */
#define E_   (B_ * EPB_)            /* 147456 */
#define NF_  ((size_t)B_ * N_ * H0_)/* 8388608 */

#if __has_builtin(__builtin_amdgcn_tensor_load_to_lds) && \
    __has_builtin(__builtin_amdgcn_s_wait_tensorcnt)
#define USE_TDM 1
#else
#define USE_TDM 0
#endif

// ---------------- per-graph MLP (4 rows, trivial) ---------------------------
__global__ __launch_bounds__(256) void mlp_kernel(
    const float* __restrict__ z,
    const float* __restrict__ W1, const float* __restrict__ b1,
    const float* __restrict__ g1, const float* __restrict__ be1,
    const float* __restrict__ rm1, const float* __restrict__ rv1,
    const float* __restrict__ W2, const float* __restrict__ b2,
    const float* __restrict__ g2, const float* __restrict__ be2,
    const float* __restrict__ rm2, const float* __restrict__ rv2,
    float* __restrict__ xbuf) {
  __shared__ float zsh[D_];
  __shared__ float x1[H1_];
  const int b = blockIdx.x;
  const int t = threadIdx.x;
  zsh[t] = z[b * D_ + t];
  __syncthreads();
  {
    const int h = t;
    float acc = b1[h];
    const float* w = W1 + (size_t)h * D_;
#pragma unroll 8
    for (int d = 0; d < D_; ++d) acc = fmaf(zsh[d], w[d], acc);
    acc = (acc - rm1[h]) * (1.0f / sqrtf(rv1[h] + 1e-5f)) * g1[h] + be1[h];
    x1[h] = acc > 0.0f ? acc : 0.2f * acc;
  }
  __syncthreads();
  for (int h = t; h < H0_; h += 256) {
    float acc = b2[h];
    const float* w = W2 + (size_t)h * H1_;
#pragma unroll 8
    for (int d = 0; d < H1_; ++d) acc = fmaf(x1[d], w[d], acc);
    acc = (acc - rm2[h]) * (1.0f / sqrtf(rv2[h] + 1e-5f)) * g2[h] + be2[h];
    xbuf[b * H0_ + h] = acc > 0.0f ? acc : 0.2f * acc;
  }
}

// ---------------- broadcast node_features (33.5 MB streaming out) -----------
__global__ __launch_bounds__(256) void bcast_kernel(
    const float* __restrict__ xbuf, float* __restrict__ nf) {
  const size_t i = (size_t)blockIdx.x * 256 + threadIdx.x;  // < B*N*H0
  const int h = (int)(i & (H0_ - 1));
  const int b = (int)(i >> 21);  // N*H0 = 2^21
  __builtin_nontemporal_store(xbuf[(b << 9) + h], &nf[i]);
}

// ---------------- zn = z + 0.1*noise ; sq = rowwise |zn|^2 ------------------
__global__ __launch_bounds__(256) void zn_kernel(
    const float* __restrict__ z, const float* __restrict__ noise,
    float* __restrict__ zn, float* __restrict__ sq) {
  const int tid = threadIdx.x;
  const int wave = tid >> 5, lane = tid & 31;
  const int row = blockIdx.x * 8 + wave;          // [0, B*N)
  const int b = row >> 12;                        // / N
  const float* zr = z + b * D_;
  const float* nr = noise + (size_t)row * D_;
  float* zo = zn + (size_t)row * D_;
  float s = 0.0f;
#pragma unroll
  for (int i = 0; i < D_ / 32; ++i) {
    const int d = lane + 32 * i;                  // coalesced
    const float v = fmaf(0.1f, nr[d], zr[d]);
    zo[d] = v;
    s = fmaf(v, v, s);
  }
#pragma unroll
  for (int off = 16; off; off >>= 1) s += __shfl_xor(s, off);  // wave32
  if (lane == 0) sq[row] = s;
}

// ---------------- fused Gram (f32 WMMA) + dist + running top-8 --------------
// Block = 8 waves = 128 rows of one batch. Column panels (16 cols x 256 f32)
// are shared block-wide in LDS, double-buffered, DMA'd by the TDM.
__global__ __launch_bounds__(256) void knn_kernel(
    const float* __restrict__ zn, const float* __restrict__ sq,
    float* __restrict__ ei_src, float* __restrict__ ei_tgt,
    float* __restrict__ ew) {
  __shared__ float Bsh[2][16][BST_];  // 33.3 KB, padded stride: no bank conflicts
  __shared__ float Dsh[8][16][17];    // per-wave 16x16 dist tile

  const int tid = threadIdx.x;
  const int wave = tid >> 5, lane = tid & 31;
  const int rt = blockIdx.x * 8 + wave;           // [0, B*(N/16))
  const int b = rt >> 8;                          // / (N/16)
  const int rowbase = (rt & 255) << 4;
  const int l16 = lane & 15;
  const int h = lane >> 4;                        // half-wave K sub-pair
  const float* zb = zn + (size_t)b * N_ * D_;
  const float* sqb = sq + b * N_;

  // A fragment source: lane holds row rowbase+l16, K pairs {k0+2h, k0+2h+1}.
  // 16 KB per wave, reused across all 256 tile iterations (WGP$-resident).
  const float* ap = zb + (size_t)(rowbase + l16) * D_ + 2 * h;

#if !USE_TDM
  // fallback staging registers for the next panel (4 x float4 per thread)
  const int fc = tid >> 4;            // column 0..15
  const int fs = tid & 15;            // 16-float segment within the row
  float4 pr[4];
#endif

  // ---- panel fill: issue (async) -------------------------------------------
  auto panel_issue = [&](int ct1, int nbuf) {
#if USE_TDM
    if (wave == 0) {
      const unsigned ldsa = (unsigned)(uintptr_t)&Bsh[nbuf][0][0];
      const unsigned long long ga =
          (unsigned long long)(uintptr_t)(zb + (size_t)(ct1 * 16) * D_);
      // D# group 0: count=1 | lds_addr | global_addr[56:0] | type=2
      v4u g0 = { 1u, ldsa, (unsigned)(ga & 0xffffffffu),
                 (unsigned)(((ga >> 32) & 0x1ffffffu) | (2u << 30)) };
      // D# group 1: data_size=4B, pad 4 DWORDs every 256 DWORDs (-> stride 260),
      // tensor_dim0=256, tensor_dim1=4096, tile=256x16, dim0_stride=256.
      v8i g1 = { (int)((2u << 16) | (1u << 20) | (7u << 22) | (3u << 25)),
                 (int)(256u << 16),                  // tensor_dim0 lo16 << 16
                 (int)(4096u << 16),                 // dim0 hi=0 | tensor_dim1 lo16
                 (int)(256u << 16),                  // dim1 hi=0 | tile_dim0=256
                 16,                                 // tile_dim1=16, tile_dim2=0
                 256,                                // tensor_dim0_stride lo32
                 0, 0 };
      v4i gz4 = { 0, 0, 0, 0 };                      // groups 2/3 unused (2D)
      v8i gz8 = { 0, 0, 0, 0, 0, 0, 0, 0 };
      __builtin_amdgcn_tensor_load_to_lds(g0, g1, gz4, gz4, gz8, 0);
    }
#else
    const float* src = zb + (size_t)(ct1 * 16 + fc) * D_ + fs * 16;
#pragma unroll
    for (int i = 0; i < 4; ++i) pr[i] = *(const float4*)(src + 4 * i);
#endif
  };

  // ---- panel fill: commit (make LDS panel valid; caller barriers) ----------
  auto panel_commit = [&](int nbuf) {
#if USE_TDM
    if (wave == 0) __builtin_amdgcn_s_wait_tensorcnt(0);
    (void)nbuf;
#else
#pragma unroll
    for (int i = 0; i < 4; ++i)
      *(float4*)&Bsh[nbuf][fc][fs * 16 + 4 * i] = pr[i];
#endif
  };

  float srow[8];
#pragma unroll
  for (int v = 0; v < 8; ++v) srow[v] = sqb[rowbase + v + 8 * h];

  float best[K_];
  int bidx[K_];
#pragma unroll
  for (int k = 0; k < K_; ++k) { best[k] = 3.0e38f; bidx[k] = 0; }

  // prologue: panel 0 into buffer 0
  panel_issue(0, 0);
  panel_commit(0);
  __syncthreads();

  for (int ct = 0; ct < NT_; ++ct) {
    const int buf = ct & 1;
    if (ct + 1 < NT_) panel_issue(ct + 1, buf ^ 1);   // DMA overlaps compute
    if (wave == 1 && ct + 2 < NT_ && lane < 16)
      __builtin_prefetch(zb + (size_t)((ct + 2) * 16 + lane) * D_, 0, 1);

    // ---- 16x16 Gram tile: 64 x v_wmma_f32_16x16x4_f32, A:global, B:LDS ----
    const float* blp = &Bsh[buf][l16][2 * h];
    v8f acc = {};
    v2f a0 = *(const v2f*)(ap);
    v2f b0 = *(const v2f*)(blp);
#pragma unroll 7
    for (int k0 = 0; k0 < D_ - 4; k0 += 4) {   // software-pipelined 1 stage
      v2f a1 = *(const v2f*)(ap + k0 + 4);
      v2f b1 = *(const v2f*)(blp + k0 + 4);
      acc = __builtin_amdgcn_wmma_f32_16x16x4_f32(
          false, a0, false, b0, (short)0, acc, false, false);
      a0 = a1; b0 = b1;
    }
    acc = __builtin_amdgcn_wmma_f32_16x16x4_f32(
        false, a0, false, b0, (short)0, acc, false, false);

    // C/D layout: VGPR v = row v (lanes 0-15) / row v+8 (lanes 16-31), col=l16
    const float scol = sqb[ct * 16 + l16];
#pragma unroll
    for (int v = 0; v < 8; ++v) {
      const float d2 = srow[v] + scol - 2.0f * acc[v];
      Dsh[wave][v + 8 * h][l16] = sqrtf(fmaxf(d2, 0.0f));
    }
    __syncthreads();                       // Dsh visible wave-wide

    if (lane < 16) {  // lane r owns row rowbase+r's running top-8
#pragma unroll 4
      for (int j = 0; j < 16; ++j) {
        const float d = Dsh[wave][lane][j];
        if (d < best[K_ - 1]) {
          float cd = d;
          int ci = ct * 16 + j;
#pragma unroll
          for (int p = 0; p < K_; ++p) {  // strict '<': lower index wins ties
            if (cd < best[p]) {
              float tf = best[p]; best[p] = cd; cd = tf;
              int ti = bidx[p]; bidx[p] = ci; ci = ti;
            }
          }
        }
      }
    }
    if (ct + 1 < NT_) panel_commit(buf ^ 1);
    __syncthreads();                       // next panel ready + Dsh reusable
  }

  if (lane < 16) {
    const int n = rowbase + lane;
    const int gsrc = b * N_ + n;
    const size_t ebase = (size_t)b * EPB_ + (size_t)n * K_;
#pragma unroll
    for (int k = 0; k < K_; ++k) {
      ei_src[ebase + k] = (float)gsrc;
      ei_tgt[ebase + k] = (float)(b * N_ + bidx[k]);
      ew[ebase + k] = 1.0f / (1.0f + best[k]);
    }
    const size_t e2 = (size_t)b * EPB_ + (size_t)(N_ * K_) + n;
    ei_src[e2] = (float)gsrc;
    ei_tgt[e2] = (float)gsrc;
    ew[e2] = 1.0f;
  }
}

// ---------------------------------------------------------------------------
extern "C" void kernel_launch(void* const* d_in, const int* in_sizes, int n_in,
                              void* d_out, int out_size, void* d_ws, size_t ws_size,
                              hipStream_t stream) {
  const float* z   = (const float*)d_in[0];
  const float* nz  = (const float*)d_in[1];
  const float* W1  = (const float*)d_in[2];
  const float* b1  = (const float*)d_in[3];
  const float* g1  = (const float*)d_in[4];
  const float* be1 = (const float*)d_in[5];
  const float* rm1 = (const float*)d_in[6];
  const float* rv1 = (const float*)d_in[7];
  const float* W2  = (const float*)d_in[8];
  const float* b2  = (const float*)d_in[9];
  const float* g2  = (const float*)d_in[10];
  const float* be2 = (const float*)d_in[11];
  const float* rm2 = (const float*)d_in[12];
  const float* rv2 = (const float*)d_in[13];
  // d_in[14] = num_nodes (4096, compiled in)

  // workspace: zn (16 MB) | sq (64 KB) | xbuf (8 KB)
  float* zn   = (float*)d_ws;
  float* sq   = zn + (size_t)B_ * N_ * D_;
  float* xbuf = sq + (size_t)B_ * N_;

  // output: node_features | edge_index(src,tgt as float) | edge_weights
  float* nf     = (float*)d_out;
  float* ei_src = nf + NF_;
  float* ei_tgt = ei_src + E_;
  float* ew     = ei_tgt + E_;

  mlp_kernel<<<B_, 256, 0, stream>>>(z, W1, b1, g1, be1, rm1, rv1,
                                     W2, b2, g2, be2, rm2, rv2, xbuf);
  zn_kernel<<<(B_ * N_) / 8, 256, 0, stream>>>(z, nz, zn, sq);
  bcast_kernel<<<(unsigned)(NF_ / 256), 256, 0, stream>>>(xbuf, nf);
  knn_kernel<<<(B_ * (N_ / 16)) / 8, 256, 0, stream>>>(zn, sq, ei_src, ei_tgt, ew);
}